// GraphConvolution_66262755443070
// MI455X (gfx1250) — compile-verified
//
#include <hip/hip_runtime.h>
#include <hip/hip_bf16.h>

typedef __attribute__((ext_vector_type(2))) float v2f;
typedef __attribute__((ext_vector_type(8))) float v8f;

#define IN_F 64
#define OUT_F 64

// ---------------------------------------------------------------------------
// 1) zero the degree array
__global__ void gcn_zero_deg(float* __restrict__ deg, int n_nodes) {
    int i = blockIdx.x * blockDim.x + threadIdx.x;
    if (i < n_nodes) deg[i] = 0.0f;
}

// 2) degree = bincount over destination (row) indices
__global__ void gcn_degree(const long long* __restrict__ ei,
                           float* __restrict__ deg, int n_edges) {
    int e = blockIdx.x * blockDim.x + threadIdx.x;
    if (e < n_edges) atomicAdd(deg + (int)ei[e], 1.0f);
}

// 3) deg -> deg^{-1/2} in place (0 where deg == 0)
__global__ void gcn_dinvsqrt(float* __restrict__ deg, int n_nodes) {
    int i = blockIdx.x * blockDim.x + threadIdx.x;
    if (i < n_nodes) {
        float d = deg[i];
        deg[i] = (d > 0.0f) ? rsqrtf(fmaxf(d, 1.0f)) : 0.0f;
    }
}

// 4) support = x @ weight  via V_WMMA_F32_16X16X4_F32.
//    W is staged in LDS with K-pairs interleaved so each lane's B fragment
//    (W[k][n], W[k+1][n]) is one contiguous 8-byte ds_load_b64.
//    LDS layout: wldsP[(k>>1)*128 + n*2 + (k&1)]
__global__ void gcn_gemm_wmma(const float* __restrict__ x,
                              const float* __restrict__ w,
                              float* __restrict__ support, int n_nodes) {
    __shared__ float wldsP[IN_F * OUT_F];   // 16 KB of the 320 KB WGP LDS
    for (int idx = threadIdx.x; idx < IN_F * OUT_F; idx += blockDim.x) {
        const int k = idx >> 6;
        const int n = idx & 63;
        wldsP[(k >> 1) * 128 + n * 2 + (k & 1)] = w[idx];
    }
    __syncthreads();

    const int wave = threadIdx.x >> 5;           // wave32
    const int lane = threadIdx.x & 31;
    const int tile = blockIdx.x * (blockDim.x >> 5) + wave;
    const int row0 = tile * 16;
    if (row0 >= n_nodes) return;                 // wave-uniform: EXEC stays all-1s

    const int m  = lane & 15;                    // M (A) / N (B,D) within tile
    const int kh = (lane >> 4) * 2;              // lane-group K offset (0 or 2)

    v8f acc[4] = {};                             // 4 x (16x16 f32) = 64 cols

    for (int k0 = 0; k0 < IN_F; k0 += 4) {
        // A fragment: x[row0+m][k0 + kh + {0,1}]  (contiguous float2)
        const float2 av = *(const float2*)(x + (size_t)(row0 + m) * IN_F + k0 + kh);
        v2f a; a.x = av.x; a.y = av.y;
        // Base of this lane's B fragments for the current K-slice
        const float* bbase = wldsP + ((k0 + kh) >> 1) * 128 + m * 2;
        #pragma unroll
        for (int nt = 0; nt < 4; ++nt) {
            const float2 bv = *(const float2*)(bbase + nt * 32);  // one ds_load_b64
            v2f b; b.x = bv.x; b.y = bv.y;
            acc[nt] = __builtin_amdgcn_wmma_f32_16x16x4_f32(
                false, a, false, b, (short)0, acc[nt], false, false);
        }
    }

    // D layout: (M = v + 8*(lane>>4), N = lane&15)
    const int mrow = row0 + 8 * (lane >> 4);
    #pragma unroll
    for (int nt = 0; nt < 4; ++nt) {
        #pragma unroll
        for (int v = 0; v < 8; ++v) {
            support[(size_t)(mrow + v) * OUT_F + nt * 16 + m] = acc[nt][v];
        }
    }
}

// 5) out[n][f] = bias[f]
__global__ void gcn_init_out(const float* __restrict__ bias,
                             float* __restrict__ out, int total) {
    int i = blockIdx.x * blockDim.x + threadIdx.x;
    if (i < total) out[i] = bias[i & (OUT_F - 1)];
}

// 6) per-edge gather-scale-scatter: one wave per edge, float2 per lane.
//    support/out both live in L2 (25.6 MB << 192 MB).
__global__ void gcn_scatter(const long long* __restrict__ ei,
                            const float* __restrict__ ew,
                            const float* __restrict__ dinv,
                            const float* __restrict__ support,
                            float* __restrict__ out, int n_edges) {
    const int gwave = (int)((blockIdx.x * blockDim.x + threadIdx.x) >> 5);
    const int lane  = threadIdx.x & 31;
    if (gwave >= n_edges) return;

    const int r = (int)ei[gwave];                    // row (destination)
    const int c = (int)ei[(size_t)n_edges + gwave];  // col (source)
    const float norm = dinv[r] * ew[gwave] * dinv[c];

    const float2 s = *(const float2*)(support + (size_t)c * OUT_F + lane * 2);
    float* o = out + (size_t)r * OUT_F + lane * 2;
    atomicAdd(o + 0, norm * s.x);
    atomicAdd(o + 1, norm * s.y);
}

// ---------------------------------------------------------------------------
extern "C" void kernel_launch(void* const* d_in, const int* in_sizes, int n_in,
                              void* d_out, int out_size, void* d_ws, size_t ws_size,
                              hipStream_t stream) {
    const float*     x    = (const float*)d_in[0];
    const long long* ei   = (const long long*)d_in[1];   // int64 [2, E]
    const float*     ew   = (const float*)d_in[2];
    const float*     w    = (const float*)d_in[3];
    const float*     bias = (const float*)d_in[4];
    float*           out  = (float*)d_out;

    const int n_nodes = in_sizes[0] / IN_F;
    const int n_edges = in_sizes[2];

    float* deg     = (float*)d_ws;                 // n_nodes floats (reused as deg^-1/2)
    float* support = (float*)d_ws + 65536;         // +256 KB: n_nodes*64 floats

    // degree + normalization
    gcn_zero_deg<<<(n_nodes + 255) / 256, 256, 0, stream>>>(deg, n_nodes);
    gcn_degree<<<(n_edges + 255) / 256, 256, 0, stream>>>(ei, deg, n_edges);
    gcn_dinvsqrt<<<(n_nodes + 255) / 256, 256, 0, stream>>>(deg, n_nodes);

    // support = x @ weight (WMMA): 16-row tile per wave, 8 waves per block
    const int tiles  = (n_nodes + 15) / 16;
    const int blocks = (tiles + 7) / 8;
    gcn_gemm_wmma<<<blocks, 256, 0, stream>>>(x, w, support, n_nodes);

    // out = bias
    gcn_init_out<<<(n_nodes * OUT_F + 255) / 256, 256, 0, stream>>>(bias, out, n_nodes * OUT_F);

    // scatter: 8 edges per 256-thread block (one wave32 per edge)
    gcn_scatter<<<(n_edges + 7) / 8, 256, 0, stream>>>(ei, ew, deg, support, out, n_edges);
}